// RegionLoss_60644938220193
// MI455X (gfx1250) — compile-verified
//
#include <hip/hip_runtime.h>
#include <hip/hip_bf16.h>
#include <math.h>

// ---------------- problem constants ----------------
#define NB       128
#define GG       26
#define NA       5
#define NC       20
#define NT       1024
#define STRIDE_F 32.0f
#define CH       (NC + 5)               // 25
#define GSQ      (GG * GG)              // 676
#define CELLS    (NB * NA * GSQ)        // 432640
#define PLANES   (NB * NA)              // 640 (one block per (b,a) plane)
#define PLANE_F  (CH * GSQ)             // 16900 floats, contiguous in x AND out
#define PLANE_CHUNKS ((PLANE_F * 4) / 16)  // 4225 x 16B
#define NCHUNK   (PLANES / 4)           // 160 WMMA accumulation steps
#define OUT_ROWS (NA * GSQ)             // 3380
#define OUT_ELEMS (NB * OUT_ROWS * CH)  // 10,816,000
#define OBJ_SCALE    5.0f
#define IGNORE_TH    0.6f

__constant__ float c_AW[NA] = {1.08f, 3.42f, 6.63f, 9.42f, 16.62f};
__constant__ float c_AH[NA] = {1.19f, 4.41f, 11.38f, 5.11f, 10.52f};

typedef float v2f __attribute__((ext_vector_type(2)));
typedef float v8f __attribute__((ext_vector_type(8)));

__device__ __forceinline__ float sigf(float v) { return 1.0f / (1.0f + __expf(-v)); }

__device__ __forceinline__ unsigned lds_off(const void* p) {
    // generic LDS-aperture address: addr[31:0] is the LDS byte offset
    return (unsigned)(unsigned long long)p;
}

// ---------------- kernel 0: zero per-cell flags ----------------
__global__ void k_init(int* __restrict__ flags) {
    int idx = blockIdx.x * blockDim.x + threadIdx.x;
    if (idx < CELLS) flags[idx] = 0;
}

// ---------------- kernel 1: per-target scatter ----------------
__global__ void k_targets(const float* __restrict__ x,
                          const float* __restrict__ tgt,
                          int* __restrict__ flags, int* __restrict__ label,
                          float* __restrict__ wtx, float* __restrict__ wty,
                          float* __restrict__ wtw, float* __restrict__ wth,
                          float* __restrict__ wiou) {
    int t = blockIdx.x * blockDim.x + threadIdx.x;
    if (t >= NT) return;
    const float* r = tgt + t * 6;
    int   b   = (int)r[0];
    int   lab = (int)r[1];
    float gx  = r[2] * (float)GG, gy = r[3] * (float)GG;
    float gw  = r[4] * (float)GG, gh = r[5] * (float)GG;
    int gi = (int)floorf(gx); if (gi < 0) gi = 0; if (gi > GG - 1) gi = GG - 1;
    int gj = (int)floorf(gy); if (gj < 0) gj = 0; if (gj > GG - 1) gj = GG - 1;

    float best = -1.0f; int bn = 0;
    float aiou[NA];
    #pragma unroll
    for (int a = 0; a < NA; ++a) {
        float inter = fminf(c_AW[a], gw) * fminf(c_AH[a], gh);
        float uni   = c_AW[a] * c_AH[a] + 1e-16f + gw * gh - inter;
        aiou[a] = inter / uni;
        if (aiou[a] > best) { best = aiou[a]; bn = a; }
    }

    int cell = ((b * NA + bn) * GG + gj) * GG + gi;
    wtx[cell] = gx - floorf(gx);
    wty[cell] = gy - floorf(gy);
    wtw[cell] = __logf(gw / c_AW[bn] + 1e-16f);
    wth[cell] = __logf(gh / c_AH[bn] + 1e-16f);
    label[cell] = lab;

    size_t base = (size_t)(b * (NA * CH) + bn * CH) * GSQ + (size_t)gj * GG + gi;
    float px = sigf(x[base]);
    float py = sigf(x[base + GSQ]);
    float pw = x[base + 2 * GSQ];
    float ph = x[base + 3 * GSQ];
    float bx = px + (float)gi, by = py + (float)gj;
    float bw = __expf(pw) * c_AW[bn], bh = __expf(ph) * c_AH[bn];

    float b1x1 = bx - bw * 0.5f, b1x2 = bx + bw * 0.5f;
    float b1y1 = by - bh * 0.5f, b1y2 = by + bh * 0.5f;
    float b2x1 = gx - gw * 0.5f, b2x2 = gx + gw * 0.5f;
    float b2y1 = gy - gh * 0.5f, b2y2 = gy + gh * 0.5f;
    float iw = fmaxf(fminf(b1x2, b2x2) - fmaxf(b1x1, b2x1) + 1.0f, 0.0f);
    float ih = fmaxf(fminf(b1y2, b2y2) - fmaxf(b1y1, b2y1) + 1.0f, 0.0f);
    float inter = iw * ih;
    float a1 = (b1x2 - b1x1 + 1.0f) * (b1y2 - b1y1 + 1.0f);
    float a2 = (b2x2 - b2x1 + 1.0f) * (b2y2 - b2y1 + 1.0f);
    wiou[cell] = inter / (a1 + a2 - inter + 1e-16f);

    atomicOr(&flags[cell], 3);   // bit0: obj, bit1: noobj cleared
    #pragma unroll
    for (int a = 0; a < NA; ++a)
        if (aiou[a] > IGNORE_TH) {
            int c2 = ((b * NA + a) * GG + gj) * GG + gi;
            atomicOr(&flags[c2], 2);
        }
}

__device__ __forceinline__ float waveRed(float v) {
    #pragma unroll
    for (int off = 16; off > 0; off >>= 1) v += __shfl_xor(v, off);
    return v;
}

// ---------------- kernel 2: async-DMA streaming forward + loss partials ----
// One block per (b, anchor) plane. Input plane and output plane are both
// contiguous 67.6 KB regions -> stage both through LDS with the async
// LDS-DMA path (ASYNCcnt), compute entirely out of LDS.
__global__ void k_main(const float* __restrict__ x,
                       const int* __restrict__ flags, const int* __restrict__ label,
                       const float* __restrict__ wtx, const float* __restrict__ wty,
                       const float* __restrict__ wtw, const float* __restrict__ wth,
                       const float* __restrict__ wiou,
                       float* __restrict__ out, float* __restrict__ partials) {
    extern __shared__ float smem[];
    float* s_in  = smem;                 // PLANE_F floats
    float* s_out = smem + PLANE_F;       // PLANE_F floats
    float* s_acc = smem + 2 * PLANE_F;   // 16 floats

    const int plane = blockIdx.x;
    const int a   = plane % NA;
    const int tid = threadIdx.x;

    // ---- stage 1: bulk async copy of the whole input plane into LDS ----
    const float* gin = x + (size_t)plane * PLANE_F;
    unsigned in_base = lds_off(s_in);
    for (int k = tid; k < PLANE_CHUNKS; k += 256) {
        unsigned long long ga = (unsigned long long)(const void*)gin
                              + (unsigned long long)k * 16ull;
        unsigned la = in_base + (unsigned)k * 16u;
        asm volatile("global_load_async_to_lds_b128 %0, %1, off"
                     :: "v"(la), "v"(ga) : "memory");
    }
    asm volatile("s_wait_asynccnt 0" ::: "memory");
    __syncthreads();

    if (tid < 16) s_acc[tid] = 0.0f;
    __syncthreads();

    // ---- stage 2: compute from LDS, write results to LDS out-plane ----
    float q[9];
    #pragma unroll
    for (int k = 0; k < 9; ++k) q[k] = 0.0f;

    for (int t = tid; t < GSQ; t += 256) {
        int i = t % GG, j = t / GG;
        int idx = plane * GSQ + t;

        float v[CH];
        #pragma unroll
        for (int c = 0; c < CH; ++c) v[c] = s_in[c * GSQ + t];

        float px = sigf(v[0]), py = sigf(v[1]);
        float pw = v[2],       ph = v[3];
        float pconf = sigf(v[4]);

        float* orow = s_out + t * CH;
        orow[0] = (px + (float)i) * STRIDE_F;
        orow[1] = (py + (float)j) * STRIDE_F;
        orow[2] = __expf(pw) * c_AW[a] * STRIDE_F;
        orow[3] = __expf(ph) * c_AH[a] * STRIDE_F;
        orow[4] = pconf;

        int   fl  = flags[idx];
        float om  = (fl & 1) ? 1.0f : 0.0f;
        float nm  = (fl & 2) ? 0.0f : 1.0f;
        int   lab = label[idx];

        float lcls = 0.0f;
        #pragma unroll
        for (int c = 0; c < NC; ++c) {
            float p = sigf(v[5 + c]);
            orow[5 + c] = p;
            if (fl & 1) {
                float pc = fminf(fmaxf(p, 1e-12f), 1.0f - 1e-12f);
                float tc = (c == lab) ? 1.0f : 0.0f;
                lcls -= tc * __logf(pc) + (1.0f - tc) * __logf(1.0f - pc);
            }
        }

        float tx = (fl & 1) ? wtx[idx] : 0.0f;
        float ty = (fl & 1) ? wty[idx] : 0.0f;
        float tw = (fl & 1) ? wtw[idx] : 0.0f;
        float th = (fl & 1) ? wth[idx] : 0.0f;
        float ti = (fl & 1) ? wiou[idx] : 0.0f;
        q[0] += om * (px - tx) * (px - tx);
        q[1] += om * (py - ty) * (py - ty);
        q[2] += om * (pw - tw) * (pw - tw);
        q[3] += om * (ph - th) * (ph - th);
        q[4] += om * (pconf - ti) * (pconf - ti);
        q[5] += nm * pconf * pconf;          // tconf == 0 on noobj cells
        q[6] += lcls;
        q[7] += om;
        q[8] += nm;
    }

    // ---- stage 3: block reduction of the 9 loss partials ----
    #pragma unroll
    for (int k = 0; k < 9; ++k) {
        float s = waveRed(q[k]);                              // wave32 butterfly
        if ((tid & 31) == 0) atomicAdd(&s_acc[k], s);         // ds_add_f32
    }
    __syncthreads();   // s_acc final AND s_out fully written
    if (tid < 16) partials[(size_t)blockIdx.x * 16 + tid] = s_acc[tid];

    // ---- stage 4: bulk async store of the output plane from LDS ----
    float* gout = out + (size_t)plane * PLANE_F;
    unsigned out_base = lds_off(s_out);
    for (int k = tid; k < PLANE_CHUNKS; k += 256) {
        unsigned long long ga = (unsigned long long)(void*)gout
                              + (unsigned long long)k * 16ull;
        unsigned la = out_base + (unsigned)k * 16u;
        asm volatile("global_store_async_from_lds_b128 %0, %1, off"
                     :: "v"(ga), "v"(la) : "memory");
    }
    asm volatile("s_wait_asynccnt 0" ::: "memory");
}

// ---------------- kernel 3: WMMA-based final reduction (1 wave) ----------------
// totals[n] = sum over PLANES rows of partials[row][n], via
// V_WMMA_F32_16X16X4_F32 with all-ones A: D[m,n] += sum_k B[k,n].
__global__ void k_finalize(const float* __restrict__ partials, float* __restrict__ loss) {
    int lane = threadIdx.x;          // 0..31, EXEC all ones, uniform loop
    int half = lane >> 4;
    int col  = lane & 15;
    v2f ones; ones[0] = 1.0f; ones[1] = 1.0f;
    v8f acc = {};
    for (int c = 0; c < NCHUNK; ++c) {
        int r0 = 4 * c + half;
        int r1 = 4 * c + 2 + half;
        v2f bm;
        bm[0] = partials[(size_t)r0 * 16 + col];
        bm[1] = partials[(size_t)r1 * 16 + col];
        acc = __builtin_amdgcn_wmma_f32_16x16x4_f32(
                  false, ones, false, bm, (short)0, acc, false, false);
    }
    __shared__ float tot[16];
    if (lane < 16) tot[lane] = acc[0];   // D row 0 (all rows identical)
    __syncthreads();
    if (lane == 0) {
        float n_obj   = tot[7];
        float n_noobj = tot[8];
        float ino = 1.0f / n_obj;
        float l = tot[0] * ino + tot[1] * ino + tot[2] * ino + tot[3] * ino;
        l += OBJ_SCALE * tot[4] * ino + tot[5] / n_noobj;
        l += tot[6] / (n_obj * (float)NC);
        loss[0] = l;
    }
}

// ---------------- launcher ----------------
extern "C" void kernel_launch(void* const* d_in, const int* in_sizes, int n_in,
                              void* d_out, int out_size, void* d_ws, size_t ws_size,
                              hipStream_t stream) {
    const float* x   = (const float*)d_in[0];
    const float* tgt = (const float*)d_in[1];
    float* out = (float*)d_out;

    int*   flags    = (int*)d_ws;
    int*   label    = flags + CELLS;
    float* wtx      = (float*)(label + CELLS);
    float* wty      = wtx + CELLS;
    float* wtw      = wty + CELLS;
    float* wth      = wtw + CELLS;
    float* wiou     = wth + CELLS;
    float* partials = wiou + CELLS;   // PLANES*16 floats (every row written)

    k_init<<<(CELLS + 255) / 256, 256, 0, stream>>>(flags);
    k_targets<<<(NT + 255) / 256, 256, 0, stream>>>(x, tgt, flags, label,
                                                    wtx, wty, wtw, wth, wiou);
    size_t shmem = (size_t)(2 * PLANE_F + 16) * sizeof(float);  // 135,264 B
    k_main<<<PLANES, 256, shmem, stream>>>(x, flags, label, wtx, wty, wtw, wth,
                                           wiou, out, partials);
    k_finalize<<<1, 32, 0, stream>>>(partials, out + OUT_ELEMS);
}